// MessagePassing_33852932227668
// MI455X (gfx1250) — compile-verified
//
#include <hip/hip_runtime.h>

typedef __attribute__((ext_vector_type(2))) float v2f;
typedef __attribute__((ext_vector_type(8))) float v8f;

#define TB   256
#define HDIM 128
#define LWS  132      // padded LDS row stride for A tile (floats)
#define PSTR 288      // pair-row stride for fragment-ready W (floats): +32 banks/row
#define BN_EPS 1e-5f

// ---------------------------------------------------------------------------
// Generic fused GEMM:  Y[n,h] = act( sum_d A'[n,d] * W[h,d] + bias[h] )
//   A'[n,d] = useBn ? BN(X[n,d], channel d) : X[n,d]
//   act = relu ? max(0,.) : identity
// One block: 128 rows x 128 cols, 8 wave32s, WMMA f32 16x16x4 (full f32).
// A tile streamed HBM->LDS with global_load_async_to_lds_b128 (ASYNCcnt).
// W stored pair-interleaved in LDS so each B fragment is one ds_load_b64.
// ---------------------------------------------------------------------------
__global__ __launch_bounds__(TB) void k_gemm(
    const float* __restrict__ X, const float* __restrict__ W,
    const float* __restrict__ bias, float* __restrict__ Y,
    int useBn, const float* __restrict__ stats,
    const float* __restrict__ bng, const float* __restrict__ bnb,
    float invCnt, int relu)
{
  extern __shared__ __align__(16) float lds[];
  float* lA  = lds;                  // [128][LWS]
  float* lWf = lds + 128 * LWS;      // [64][PSTR]: lWf[p][2n+j] = W[n][2p+j]

  const int t = threadIdx.x;
  const long rowBase = (long)blockIdx.x * 128;

  // Stage W pair-interleaved: fragment-ready for WMMA B operand.
  for (int i = t; i < HDIM * HDIM; i += TB) {
    int h = i >> 7, d = i & 127;                 // W[h][d]
    lWf[(d >> 1) * PSTR + h * 2 + (d & 1)] = W[i];
  }

  // Stage 128x128 A tile (tile spans whole rows -> one contiguous 64KB blob).
  if (!useBn) {
    // CDNA5 async copy: HBM -> LDS directly, tracked by ASYNCcnt.
    const float* base = X + rowBase * HDIM;      // uniform -> SGPR pair
    for (int i = t * 4; i < HDIM * HDIM; i += TB * 4) {
      int r = i >> 7, c = i & 127;
      unsigned gByte = (unsigned)i * 4u;
      unsigned lByte = (unsigned)(uintptr_t)(lA + r * LWS + c);  // low 32 = LDS addr
      asm volatile("global_load_async_to_lds_b128 %0, %1, %2"
                   :: "v"(lByte), "v"(gByte), "s"(base) : "memory");
    }
    asm volatile("s_wait_asynccnt 0" ::: "memory");
  } else {
    // BN-apply in flight (element transform -> VALU staging path).
    for (int i = t * 4; i < HDIM * HDIM; i += TB * 4) {
      int r = i >> 7, c = i & 127;
      float4 v = *(const float4*)(X + (rowBase + r) * HDIM + c);
      float* pv = (float*)&v;
#pragma unroll
      for (int j = 0; j < 4; ++j) {
        int ch = c + j;
        float mu = stats[ch] * invCnt;
        float va = stats[HDIM + ch] * invCnt - mu * mu;
        pv[j] = (pv[j] - mu) * rsqrtf(va + BN_EPS) * bng[ch] + bnb[ch];
      }
      *(float4*)(lA + r * LWS + c) = v;
    }
  }
  __syncthreads();

  const int wave = t >> 5;          // 0..7 -> 16-row strip
  const int lane = t & 31;
  const int lo = lane & 15;         // N (col) / M (row) index inside frag
  const int hi = lane >> 4;         // K-pair selector

  v8f acc[8];
#pragma unroll
  for (int n = 0; n < 8; ++n) {
#pragma unroll
    for (int v = 0; v < 8; ++v) acc[n][v] = 0.0f;
  }

  const float* aRow = lA + (wave * 16 + lo) * LWS;   // A frag: M=lo, K pair by hi
  for (int k = 0; k < HDIM; k += 4) {
    v2f a = *(const v2f*)(aRow + k + 2 * hi);        // A[M][k+2hi..+1]
    const float* wrow = lWf + ((k >> 1) + hi) * PSTR + lo * 2;
#pragma unroll
    for (int n = 0; n < 8; ++n) {
      v2f b = *(const v2f*)(wrow + n * 32);          // {W[N][k+2hi], W[N][k+2hi+1]}
      acc[n] = __builtin_amdgcn_wmma_f32_16x16x4_f32(
          false, a, false, b, (short)0, acc[n], false, false);
    }
  }

  // Epilogue: C/D layout -> lane lo = col, VGPR v = row v (+8 if hi)
#pragma unroll
  for (int n = 0; n < 8; ++n) {
    int col = n * 16 + lo;
    float bv = bias[col];
    float* yp = Y + (rowBase + wave * 16 + hi * 8) * HDIM + col;
#pragma unroll
    for (int v = 0; v < 8; ++v) {
      float y = acc[n][v] + bv;
      if (relu) y = fmaxf(y, 0.0f);
      yp[v * HDIM] = y;
    }
  }
}

// ---------------------------------------------------------------------------
// Zero a float buffer.
// ---------------------------------------------------------------------------
__global__ void k_zero(float* __restrict__ p, long n) {
  long i = (long)blockIdx.x * blockDim.x + threadIdx.x;
  long stride = (long)gridDim.x * blockDim.x;
  for (; i < n; i += stride) p[i] = 0.0f;
}

// ---------------------------------------------------------------------------
// Segment counts: cnt[dst[e]] += 1
// ---------------------------------------------------------------------------
__global__ void k_count(float* __restrict__ cnt, const int* __restrict__ dst, int E) {
  int e = blockIdx.x * blockDim.x + threadIdx.x;
  if (e < E) atomicAdd(&cnt[dst[e]], 1.0f);
}

// ---------------------------------------------------------------------------
// Fused scatter-mean accumulate: out[b,dst[e],:] += in[b,src[e],:] / max(cnt,1)
// grid = (E, B), block = 128 lanes (one per channel).
// ---------------------------------------------------------------------------
__global__ __launch_bounds__(HDIM) void k_scatter(
    float* __restrict__ out, const float* __restrict__ in,
    const int* __restrict__ src, const int* __restrict__ dst,
    const float* __restrict__ cnt, int srcN, int dstN)
{
  int e = blockIdx.x, b = blockIdx.y, h = threadIdx.x;
  int s = src[e], d = dst[e];
  float scale = 1.0f / fmaxf(cnt[d], 1.0f);
  float v = in[((long)b * srcN + s) * HDIM + h] * scale;
  atomicAdd(&out[((long)b * dstN + d) * HDIM + h], v);
}

// ---------------------------------------------------------------------------
// BN statistics: stats[h] = sum_n x[n,h]; stats[128+h] = sum_n x[n,h]^2
// ---------------------------------------------------------------------------
__global__ __launch_bounds__(HDIM) void k_bnstats(
    const float* __restrict__ x, long nRows, float* __restrict__ stats)
{
  int h = threadIdx.x;
  float s = 0.0f, s2 = 0.0f;
  for (long r = blockIdx.x; r < nRows; r += gridDim.x) {
    float v = x[r * HDIM + h];
    s += v;
    s2 += v * v;
  }
  atomicAdd(&stats[h], s);
  atomicAdd(&stats[HDIM + h], s2);
}

// ---------------------------------------------------------------------------
extern "C" void kernel_launch(void* const* d_in, const int* in_sizes, int n_in,
                              void* d_out, int out_size, void* d_ws, size_t ws_size,
                              hipStream_t stream) {
  const int B = 8, M = 20000, R = 50000, E = 100000;
  const long BM = (long)B * M;            // 160000 = 1250 * 128
  const long BR = (long)B * R;            // 400000 = 3125 * 128
  const long BMH = BM * HDIM;
  const long BRH = BR * HDIM;

  const float* nodes = (const float*)d_in[0];
  const int* r2e = (const int*)d_in[1];   // [0]=src in M, [1]=dst in R
  const int* p2e = (const int*)d_in[2];
  const int* e2r = (const int*)d_in[3];   // [0]=src in R, [1]=dst in M
  const int* e2p = (const int*)d_in[4];
  const float* w_rct = (const float*)d_in[5];
  const float* b_rct = (const float*)d_in[6];
  const float* w_prd = (const float*)d_in[7];
  const float* b_prd = (const float*)d_in[8];
  const float* w_ra1 = (const float*)d_in[9];
  const float* b_ra1 = (const float*)d_in[10];
  const float* bn_r_g = (const float*)d_in[11];
  const float* bn_r_b = (const float*)d_in[12];
  const float* w_ra2 = (const float*)d_in[13];
  const float* b_ra2 = (const float*)d_in[14];
  const float* w_r2rct = (const float*)d_in[15];
  const float* b_r2rct = (const float*)d_in[16];
  const float* w_r2prd = (const float*)d_in[17];
  const float* b_r2prd = (const float*)d_in[18];
  const float* w_ma1 = (const float*)d_in[19];
  const float* b_ma1 = (const float*)d_in[20];
  const float* bn_m_g = (const float*)d_in[21];
  const float* bn_m_b = (const float*)d_in[22];
  const float* w_ma2 = (const float*)d_in[23];
  const float* b_ma2 = (const float*)d_in[24];

  float* ws = (float*)d_ws;
  float* bufM1 = ws;                 // B*M*H  (h_rct, later aggM)
  float* bufM2 = bufM1 + BMH;        // B*M*H  (h_prd, later tM)
  float* bufR1 = bufM2 + BMH;        // B*R*H  (aggR, later reaction_hidden)
  float* bufR2 = bufR1 + BRH;        // B*R*H  (tR, later hr2rct / hr2prd)
  float* cnt1 = bufR2 + BRH;         // R
  float* cnt2 = cnt1 + R;            // R
  float* stats = cnt2 + R;           // 256
  float* outP = (float*)d_out;       // B*M*H

  const size_t ldsBytes = (size_t)(128 * LWS + 64 * PSTR) * sizeof(float); // 138 KB
  const dim3 gM((unsigned)(BM / 128)), gR((unsigned)(BR / 128));
  const dim3 gE(E, B);
  const int zb = 2048, cb = (E + TB - 1) / TB;

  // ---- node -> edge linears (f32 WMMA) ----
  k_gemm<<<gM, TB, ldsBytes, stream>>>(nodes, w_rct, b_rct, bufM1, 0, nullptr, nullptr, nullptr, 0.f, 0);
  k_gemm<<<gM, TB, ldsBytes, stream>>>(nodes, w_prd, b_prd, bufM2, 0, nullptr, nullptr, nullptr, 0.f, 0);

  // ---- agg = scatter_mean(rct, r2e) + scatter_mean(prd, p2e) into R ----
  k_zero<<<zb, TB, 0, stream>>>(bufR1, BRH);
  k_zero<<<zb, TB, 0, stream>>>(cnt1, R);
  k_zero<<<zb, TB, 0, stream>>>(cnt2, R);
  k_count<<<cb, TB, 0, stream>>>(cnt1, r2e + E, E);
  k_count<<<cb, TB, 0, stream>>>(cnt2, p2e + E, E);
  k_scatter<<<gE, HDIM, 0, stream>>>(bufR1, bufM1, r2e, r2e + E, cnt1, M, R);
  k_scatter<<<gE, HDIM, 0, stream>>>(bufR1, bufM2, p2e, p2e + E, cnt2, M, R);

  // ---- reaction tower: lin+relu -> BN stats -> (BN-apply fused) lin ----
  k_gemm<<<gR, TB, ldsBytes, stream>>>(bufR1, w_ra1, b_ra1, bufR2, 0, nullptr, nullptr, nullptr, 0.f, 1);
  k_zero<<<1, TB, 0, stream>>>(stats, 2 * HDIM);
  k_bnstats<<<1024, HDIM, 0, stream>>>(bufR2, BR, stats);
  k_gemm<<<gR, TB, ldsBytes, stream>>>(bufR2, w_ra2, b_ra2, bufR1, 1, stats, bn_r_g, bn_r_b, 1.0f / (float)BR, 0);

  // ---- reaction -> node linears + scatter back into M ----
  k_zero<<<zb, TB, 0, stream>>>(bufM1, BMH);
  k_zero<<<zb, TB, 0, stream>>>(cnt1, M);
  k_zero<<<zb, TB, 0, stream>>>(cnt2, M);
  k_count<<<cb, TB, 0, stream>>>(cnt1, e2r + E, E);
  k_count<<<cb, TB, 0, stream>>>(cnt2, e2p + E, E);
  k_gemm<<<gR, TB, ldsBytes, stream>>>(bufR1, w_r2rct, b_r2rct, bufR2, 0, nullptr, nullptr, nullptr, 0.f, 0);
  k_scatter<<<gE, HDIM, 0, stream>>>(bufM1, bufR2, e2r, e2r + E, cnt1, R, M);
  k_gemm<<<gR, TB, ldsBytes, stream>>>(bufR1, w_r2prd, b_r2prd, bufR2, 0, nullptr, nullptr, nullptr, 0.f, 0);
  k_scatter<<<gE, HDIM, 0, stream>>>(bufM1, bufR2, e2p, e2p + E, cnt2, R, M);

  // ---- node tower: lin+relu -> BN stats -> (BN-apply fused) final lin ----
  k_gemm<<<gM, TB, ldsBytes, stream>>>(bufM1, w_ma1, b_ma1, bufM2, 0, nullptr, nullptr, nullptr, 0.f, 1);
  k_zero<<<1, TB, 0, stream>>>(stats, 2 * HDIM);
  k_bnstats<<<1024, HDIM, 0, stream>>>(bufM2, BM, stats);
  k_gemm<<<gM, TB, ldsBytes, stream>>>(bufM2, w_ma2, b_ma2, outP, 1, stats, bn_m_g, bn_m_b, 1.0f / (float)BM, 0);
}